// GATv2Backbone_44805098832145
// MI455X (gfx1250) — compile-verified
//
#include <hip/hip_runtime.h>
#include <hip/hip_bf16.h>
#include <math.h>

// GATv2 backbone for MI455X (gfx1250, wave32).
// Dense node transforms via V_WMMA_F32_16X16X4_F32 (exact fp32 WMMA);
// sparse softmax-aggregation via global_atomic_add_f32 / CAS float-max.

#define HEADS 4
#define KDIM 64
#define NEG_SLOPE 0.2f
#define EPSF 1e-16f

typedef float v2f __attribute__((ext_vector_type(2)));
typedef float v8f __attribute__((ext_vector_type(8)));

// ---------------------------------------------------------------------------
// out[N,M] = A[N,K] @ W[K,M] + bias[M]   (row-major, fp32, exact)
// One wave computes one 16x16 output tile with K-steps of 4 using
// v_wmma_f32_16x16x4_f32. N must be a multiple of 16 (50000 = 3125*16).
// A-operand layout (ISA 7.12.2, 32-bit A 16x4): lanes 0-15 hold K,K+1 for
// row M=lane; lanes 16-31 hold K+2,K+3. B mirrored; C/D: VGPR i holds
// row tm+i (lanes 0-15) / tm+8+i (lanes 16-31), col tn+(lane&15).
// ---------------------------------------------------------------------------
__global__ void gat_wmma_gemm_bias(const float* __restrict__ A,
                                   const float* __restrict__ W,
                                   const float* __restrict__ bias,
                                   float* __restrict__ out,
                                   int Nrows, int K, int M) {
  const int lane = threadIdx.x & 31;
  const int wave = threadIdx.x >> 5;
  const int tiles_n = M >> 4;
  const int total = (Nrows >> 4) * tiles_n;
  const int tile = blockIdx.x * (blockDim.x >> 5) + wave;
  if (tile >= total) return;  // wave-uniform: EXEC stays all-ones for WMMA
  const int tm = (tile / tiles_n) << 4;
  const int tn = (tile % tiles_n) << 4;
  const int half = lane >> 4;  // 0 -> K,K+1 ; 1 -> K+2,K+3
  const int l15 = lane & 15;

  const float* arow = A + (size_t)(tm + l15) * K + 2 * half;
  const float* bcol = W + (size_t)(2 * half) * M + tn + l15;

  v8f acc = {};
  for (int k = 0; k < K; k += 4) {
    v2f a, b;
    a.x = arow[0];
    a.y = arow[1];
    b.x = bcol[0];
    b.y = bcol[M];
    acc = __builtin_amdgcn_wmma_f32_16x16x4_f32(false, a, false, b, (short)0,
                                                acc, false, false);
    arow += 4;
    bcol += (size_t)4 * M;
  }

  const int col = tn + l15;
  const float bv = bias[col];
  float* orow = out + (size_t)(tm + 8 * half) * M + col;
#pragma unroll
  for (int i = 0; i < 8; ++i) orow[(size_t)i * M] = acc[i] + bv;
}

// ---------------------------------------------------------------------------
// CAS-based float atomic max (exact segment_max semantics incl. negatives)
// ---------------------------------------------------------------------------
__device__ inline void atomicMaxF(float* addr, float val) {
  int* ai = (int*)addr;
  int old = *ai;
  while (__int_as_float(old) < val) {
    int assumed = old;
    old = atomicCAS(ai, assumed, __float_as_int(val));
    if (old == assumed) break;
  }
}

__global__ void gat_fill_f32(float* __restrict__ p, float v, long long n) {
  long long i = (long long)blockIdx.x * blockDim.x + threadIdx.x;
  if (i < n) p[i] = v;
}

// ---------------------------------------------------------------------------
// Per (edge, head): alpha = att . leaky_relu(xl[src]+xr[dst]); segment max.
// Self-loop edges (e >= E) synthesized in-kernel: src = dst = e - E.
// ---------------------------------------------------------------------------
template <int C>
__global__ void gat_edge_alpha_max(const int* __restrict__ ei, int E, int N,
                                   const float* __restrict__ xl,
                                   const float* __restrict__ xr,
                                   const float* __restrict__ att,
                                   float* __restrict__ alpha,
                                   float* __restrict__ amax) {
  int idx = blockIdx.x * blockDim.x + threadIdx.x;
  int tot = (E + N) * HEADS;
  if (idx >= tot) return;
  int e = idx >> 2, h = idx & 3;
  int s, d;
  if (e < E) {
    s = ei[e];
    d = ei[E + e];
  } else {
    s = d = e - E;
  }
  const float* pl = xl + ((size_t)s * HEADS + h) * C;
  const float* pr = xr + ((size_t)d * HEADS + h) * C;
  const float* pa = att + h * C;
  float acc = 0.f;
#pragma unroll
  for (int c = 0; c < C; ++c) {
    float v = pl[c] + pr[c];
    v = v > 0.f ? v : NEG_SLOPE * v;
    acc = fmaf(pa[c], v, acc);
  }
  alpha[(size_t)e * HEADS + h] = acc;
  atomicMaxF(&amax[d * HEADS + h], acc);
}

// Per (edge, head): ealpha = exp(alpha - amax[dst]); segment-sum denominator.
__global__ void gat_edge_exp_denom(const int* __restrict__ ei, int E, int N,
                                   const float* __restrict__ amax,
                                   float* __restrict__ alpha,
                                   float* __restrict__ denom) {
  int idx = blockIdx.x * blockDim.x + threadIdx.x;
  int tot = (E + N) * HEADS;
  if (idx >= tot) return;
  int e = idx >> 2, h = idx & 3;
  int d = (e < E) ? ei[E + e] : (e - E);
  float ea = expf(alpha[(size_t)e * HEADS + h] - amax[d * HEADS + h]);
  alpha[(size_t)e * HEADS + h] = ea;
  atomicAdd(&denom[d * HEADS + h], ea);
}

// Per (edge, head): aggr[dst] += xl[src] * (ealpha / (denom[dst] + eps))
template <int C>
__global__ void gat_edge_aggregate(const int* __restrict__ ei, int E, int N,
                                   const float* __restrict__ xl,
                                   const float* __restrict__ alpha,
                                   const float* __restrict__ denom,
                                   float* __restrict__ aggr) {
  int idx = blockIdx.x * blockDim.x + threadIdx.x;
  int tot = (E + N) * HEADS;
  if (idx >= tot) return;
  int e = idx >> 2, h = idx & 3;
  int s, d;
  if (e < E) {
    s = ei[e];
    d = ei[E + e];
  } else {
    s = d = e - E;
  }
  float w = alpha[(size_t)e * HEADS + h] / (denom[d * HEADS + h] + EPSF);
  const float* pl = xl + ((size_t)s * HEADS + h) * C;
  float* po = aggr + ((size_t)d * HEADS + h) * C;
#pragma unroll
  for (int c = 0; c < C; ++c) atomicAdd(&po[c], pl[c] * w);
}

// Concat layers: out = elu(aggr + bias)
__global__ void gat_elu_bias(const float* __restrict__ aggr,
                             const float* __restrict__ bias,
                             float* __restrict__ out, long long total, int F) {
  long long i = (long long)blockIdx.x * blockDim.x + threadIdx.x;
  if (i >= total) return;
  float v = aggr[i] + bias[(int)(i % F)];
  out[i] = v > 0.f ? v : (expf(v) - 1.f);
}

// Final layer: mean over heads (H=4, C=64) + bias
__global__ void gat_mean_bias(const float* __restrict__ aggr,
                              const float* __restrict__ bias,
                              float* __restrict__ out, int N) {
  int i = blockIdx.x * blockDim.x + threadIdx.x;
  if (i >= N * 64) return;
  int n = i >> 6, c = i & 63;
  const float* p = aggr + (size_t)n * 256 + c;
  out[i] = 0.25f * (p[0] + p[64] + p[128] + p[192]) + bias[c];
}

// ---------------------------------------------------------------------------

extern "C" void kernel_launch(void* const* d_in, const int* in_sizes, int n_in,
                              void* d_out, int out_size, void* d_ws,
                              size_t ws_size, hipStream_t stream) {
  (void)n_in;
  (void)out_size;
  (void)ws_size;
  const float* x = (const float*)d_in[0];
  const int* ei = (const int*)d_in[1];
  const float* Wl[3] = {(const float*)d_in[2], (const float*)d_in[8],
                        (const float*)d_in[14]};
  const float* bl[3] = {(const float*)d_in[3], (const float*)d_in[9],
                        (const float*)d_in[15]};
  const float* Wr[3] = {(const float*)d_in[4], (const float*)d_in[10],
                        (const float*)d_in[16]};
  const float* br[3] = {(const float*)d_in[5], (const float*)d_in[11],
                        (const float*)d_in[17]};
  const float* att[3] = {(const float*)d_in[6], (const float*)d_in[12],
                         (const float*)d_in[18]};
  const float* bias[3] = {(const float*)d_in[7], (const float*)d_in[13],
                          (const float*)d_in[19]};

  const int N = in_sizes[0] / KDIM;  // 50000
  const int E = in_sizes[1] / 2;     // 400000
  const int Etot = E + N;            // with self loops

  // Workspace layout (floats)
  float* ws = (float*)d_ws;
  size_t off = 0;
  float* xl = ws + off;    off += (size_t)N * 256;
  float* xr = ws + off;    off += (size_t)N * 256;
  float* aggr = ws + off;  off += (size_t)N * 256;
  float* hbuf = ws + off;  off += (size_t)N * 64;
  float* alpha = ws + off; off += (size_t)Etot * HEADS;
  float* amax = ws + off;  off += (size_t)N * HEADS;
  float* denom = ws + off; off += (size_t)N * HEADS;

  const int eth = Etot * HEADS;
  const int eblk = (eth + 255) / 256;

  for (int layer = 0; layer < 3; ++layer) {
    const int C = (layer == 2) ? 64 : 16;
    const int M = HEADS * C;
    const float* hin = (layer == 0) ? x : hbuf;

    // Node transforms: xl = hin @ Wl + bl ; xr = hin @ Wr + br   (WMMA fp32)
    const int tiles = (N / 16) * (M / 16);
    const int gblk = (tiles + 7) / 8;  // 8 waves (wave32) per 256-thread block
    gat_wmma_gemm_bias<<<gblk, 256, 0, stream>>>(hin, Wl[layer], bl[layer], xl,
                                                 N, KDIM, M);
    gat_wmma_gemm_bias<<<gblk, 256, 0, stream>>>(hin, Wr[layer], br[layer], xr,
                                                 N, KDIM, M);

    // Init segment buffers
    {
      long long n4 = (long long)N * HEADS;
      gat_fill_f32<<<(int)((n4 + 255) / 256), 256, 0, stream>>>(amax, -INFINITY,
                                                                n4);
      gat_fill_f32<<<(int)((n4 + 255) / 256), 256, 0, stream>>>(denom, 0.f, n4);
      long long nm = (long long)N * M;
      gat_fill_f32<<<(int)((nm + 255) / 256), 256, 0, stream>>>(aggr, 0.f, nm);
    }

    // Edge phase: score + segment max, softmax denom, weighted scatter-add
    if (C == 16) {
      gat_edge_alpha_max<16><<<eblk, 256, 0, stream>>>(ei, E, N, xl, xr,
                                                       att[layer], alpha, amax);
      gat_edge_exp_denom<<<eblk, 256, 0, stream>>>(ei, E, N, amax, alpha,
                                                   denom);
      gat_edge_aggregate<16><<<eblk, 256, 0, stream>>>(ei, E, N, xl, alpha,
                                                       denom, aggr);
    } else {
      gat_edge_alpha_max<64><<<eblk, 256, 0, stream>>>(ei, E, N, xl, xr,
                                                       att[layer], alpha, amax);
      gat_edge_exp_denom<<<eblk, 256, 0, stream>>>(ei, E, N, amax, alpha,
                                                   denom);
      gat_edge_aggregate<64><<<eblk, 256, 0, stream>>>(ei, E, N, xl, alpha,
                                                       denom, aggr);
    }

    // Activation / reduction
    if (layer < 2) {
      long long nm = (long long)N * M;
      gat_elu_bias<<<(int)((nm + 255) / 256), 256, 0, stream>>>(
          aggr, bias[layer], hbuf, nm, M);
    } else {
      gat_mean_bias<<<(N * 64 + 255) / 256, 256, 0, stream>>>(
          aggr, bias[2], (float*)d_out, N);
    }
  }
}